// HBV_24163486007849
// MI455X (gfx1250) — compile-verified
//
#include <hip/hip_runtime.h>
#include <stdint.h>

// HBV RHS, B ~= 2e6 independent elements. Memory-bound:
// 100 B/elem * 2e6 = 200 MB @ 23.3 TB/s => ~8.6 us floor.
//
// gfx1250 strategy:
//  - stage y (20 B/elem) and climate (12 B/elem) tiles into LDS with
//    global_load_async_to_lds_b128 (contiguous, 16B-aligned, ASYNCcnt-tracked)
//  - s_wait_asynccnt 0 + s_barrier, then bank-conflict-free LDS reads
//    (dword strides 5 and 3 are coprime with 64 banks)
//  - direct b128 stores for the 16B-aligned flux rows, dword stores for dS
//  - global_prefetch_b8 for the block's next tile

#define TILE 256

struct HBVParams {
    float tt, tti, ttm, cfr, cfmax, whc, cflux, fc, lp, beta, k0, alpha, perc, k1, dt;
};

__device__ __forceinline__ void hbv_core(const HBVParams& p,
                                         float S1, float S2, float S3, float S4, float S5,
                                         float P, float Ep, float T,
                                         float ds[5], float fx[12])
{
    const float r = 0.01f;
    const float e = 5.0f;

    const float flux_sf   = fminf(P, fmaxf(0.0f, P * (p.tt + 0.5f * p.tti - T) / p.tti));
    const float flux_refr = fmaxf(fminf(p.cfr * p.cfmax * (p.ttm - T), S2 / p.dt), 0.0f);
    const float flux_melt = fmaxf(fminf(p.cfmax * (T - p.ttm), S1 / p.dt), 0.0f);
    const float flux_rf   = fminf(P, fmaxf(0.0f, P * (T - (p.tt - 0.5f * p.tti)) / p.tti));

    const float Smax_raw = p.whc * S1;
    const float Smax     = fmaxf(Smax_raw, 0.0f);
    const float logistic = 1.0f /
        (1.0f + expf((S2 - Smax + r * e * Smax) / fmaxf(r, r * Smax)));
    const float flux_in  = (flux_rf + flux_melt) * (1.0f - logistic);
    const float flux_se  = fmaxf((S2 - Smax_raw) / p.dt, 0.0f);

    const float flux_cf  = fminf(p.cflux * (1.0f - S3 / p.fc), S4 / p.dt);
    const float flux_ea  = fminf(fminf(S3 / (p.lp * p.fc) * Ep, Ep), S3 / p.dt);
    const float flux_r   = (flux_in + flux_se) * powf(fmaxf(S3, 0.0f) / p.fc, p.beta);

    const float flux_q0   = fminf(p.k0 * powf(fmaxf(S4, 0.0f), 1.0f + p.alpha),
                                  fmaxf(S4 / p.dt, 0.0f));
    const float flux_perc = fminf(p.perc, S4 / p.dt);
    const float flux_q1   = p.k1 * S5;

    ds[0] = flux_sf + flux_refr - flux_melt;
    ds[1] = flux_rf + flux_melt - flux_refr - flux_in - flux_se;
    ds[2] = flux_in + flux_se + flux_cf - flux_ea - flux_r;
    ds[3] = flux_r - flux_cf - flux_q0 - flux_perc;
    ds[4] = flux_perc - flux_q1;

    fx[0] = flux_sf;  fx[1] = flux_refr; fx[2]  = flux_melt; fx[3]  = flux_rf;
    fx[4] = flux_in;  fx[5] = flux_se;   fx[6]  = flux_cf;   fx[7]  = flux_ea;
    fx[8] = flux_r;   fx[9] = flux_q0;   fx[10] = flux_perc; fx[11] = flux_q1;
}

__device__ __forceinline__ void async_copy_b128(const float* gsrc, const float* lds_dst)
{
    // VDST = LDS byte offset VGPR (flat LDS addr low 32 bits), VADDR = 64-bit global addr.
    const uint32_t lds_off = (uint32_t)(uintptr_t)lds_dst;
    asm volatile("global_load_async_to_lds_b128 %0, %1, off"
                 :: "v"(lds_off), "v"(gsrc)
                 : "memory");
}

__device__ __forceinline__ void store_outputs(float* __restrict__ dS_out,
                                              float* __restrict__ fx_out,
                                              int i, const float ds[5], const float fx[12])
{
    float* __restrict__ dsrow = dS_out + (size_t)5u * (size_t)i;  // 20 B, dword aligned
    dsrow[0] = ds[0]; dsrow[1] = ds[1]; dsrow[2] = ds[2]; dsrow[3] = ds[3]; dsrow[4] = ds[4];

    // 48 B at offset 48*i -> always 16 B aligned -> 3x global_store_b128
    float4* __restrict__ fxrow = (float4*)(fx_out + (size_t)12u * (size_t)i);
    fxrow[0] = make_float4(fx[0], fx[1], fx[2],  fx[3]);
    fxrow[1] = make_float4(fx[4], fx[5], fx[6],  fx[7]);
    fxrow[2] = make_float4(fx[8], fx[9], fx[10], fx[11]);
}

__global__ __launch_bounds__(TILE) void hbv_rhs_kernel(
    const float* __restrict__ y,            // (B,5)
    const float* __restrict__ theta,        // (14,)
    const float* __restrict__ climate_data, // (T,B,3)
    const float* __restrict__ dt_ptr,
    const int*   __restrict__ t_ptr,
    float* __restrict__ out,                // dS (B,5) then fluxes (B,12)
    int B)
{
    __shared__ __align__(16) float s_y[TILE * 5]; // 5120 B = 320 x 16B chunks
    __shared__ __align__(16) float s_c[TILE * 3]; // 3072 B = 192 x 16B chunks

    HBVParams p;
    p.tt    = theta[0];  p.tti   = theta[1];  p.ttm  = theta[2];  p.cfr   = theta[3];
    p.cfmax = theta[4];  p.whc   = theta[5];  p.cflux= theta[6];  p.fc    = theta[7];
    p.lp    = theta[8];  p.beta  = theta[9];  p.k0   = theta[10]; p.alpha = theta[11];
    p.perc  = theta[12]; p.k1    = theta[13]; p.dt   = dt_ptr[0];
    const int t = t_ptr[0];

    const float* __restrict__ clim = climate_data + (size_t)t * (size_t)B * 3u;
    float* __restrict__ dS_out = out;
    float* __restrict__ fx_out = out + (size_t)B * 5u;

    const int tid    = (int)threadIdx.x;
    const int ntiles = (B + TILE - 1) / TILE;
    const int nfull  = B / TILE;            // tiles with all TILE elements valid

    for (int tile = (int)blockIdx.x; tile < ntiles; tile += (int)gridDim.x) {
        const int base = tile * TILE;

        if (tile < nfull) {
            // Protect LDS from the previous iteration's readers before overwrite.
            __syncthreads();

            const float* ysrc = y    + (size_t)base * 5u;  // 16B aligned (5120*tile)
            const float* csrc = clim + (size_t)base * 3u;  // 16B aligned (3072*tile)

            // y tile: 320 chunks over 256 threads
            async_copy_b128(ysrc + 4 * tid, &s_y[4 * tid]);
            if (tid < TILE * 5 / 4 - TILE) // 64 extra chunks
                async_copy_b128(ysrc + 4 * (TILE + tid), &s_y[4 * (TILE + tid)]);
            // climate tile: 192 chunks
            if (tid < TILE * 3 / 4)
                async_copy_b128(csrc + 4 * tid, &s_c[4 * tid]);

            // Prefetch this block's next tile while current one computes.
            const int pn = tile + (int)gridDim.x;
            if (pn < nfull) {
                __builtin_prefetch(&y[(size_t)(pn * TILE + tid) * 5u], 0, 0);
                __builtin_prefetch(&clim[(size_t)(pn * TILE + tid) * 3u], 0, 0);
            }

            asm volatile("s_wait_asynccnt 0x0" ::: "memory");
            __syncthreads();

            // Bank-conflict-free LDS reads (strides 5 and 3 coprime with 64 banks).
            const float S1 = s_y[5 * tid + 0];
            const float S2 = s_y[5 * tid + 1];
            const float S3 = s_y[5 * tid + 2];
            const float S4 = s_y[5 * tid + 3];
            const float S5 = s_y[5 * tid + 4];
            const float P  = s_c[3 * tid + 0];
            const float Ep = s_c[3 * tid + 1];
            const float T  = s_c[3 * tid + 2];

            float ds[5], fx[12];
            hbv_core(p, S1, S2, S3, S4, S5, P, Ep, T, ds, fx);
            store_outputs(dS_out, fx_out, base + tid, ds, fx);
        } else {
            // Ragged tail tile: guarded direct loads (no OOB async copies).
            const int i = base + tid;
            if (i < B) {
                const size_t yb = (size_t)5u * (size_t)i;
                const size_t cb = (size_t)3u * (size_t)i;
                float ds[5], fx[12];
                hbv_core(p, y[yb], y[yb + 1], y[yb + 2], y[yb + 3], y[yb + 4],
                         clim[cb], clim[cb + 1], clim[cb + 2], ds, fx);
                store_outputs(dS_out, fx_out, i, ds, fx);
            }
        }
    }
}

extern "C" void kernel_launch(void* const* d_in, const int* in_sizes, int n_in,
                              void* d_out, int out_size, void* d_ws, size_t ws_size,
                              hipStream_t stream) {
    const float* y     = (const float*)d_in[0];
    const float* theta = (const float*)d_in[1];
    const float* clim  = (const float*)d_in[2];
    const float* dt    = (const float*)d_in[3];
    const int*   t     = (const int*)d_in[4];
    float*       out   = (float*)d_out;

    const int B = in_sizes[0] / 5;

    const int ntiles = (B + TILE - 1) / TILE;
    int blocks = ntiles < 4096 ? ntiles : 4096;  // >=2 tiles/block for prefetch overlap
    if (blocks < 1) blocks = 1;

    hbv_rhs_kernel<<<blocks, TILE, 0, stream>>>(y, theta, clim, dt, t, out, B);
}